// GAT_39977555591300
// MI455X (gfx1250) — compile-verified
//
#include <hip/hip_runtime.h>
#include <math.h>

typedef __attribute__((ext_vector_type(2))) float v2f;
typedef __attribute__((ext_vector_type(8))) float v8f;

#define HC 128
#define HEADS 8
#define HIDDEN 16
#define NEG_SLOPE 0.2f

// ---------------------------------------------------------------------------
// WMMA GEMM: C[nrows x 128] = A[nrows x 128] @ B[128 x 128], fp32 via
// V_WMMA_F32_16X16X4_F32. One block = 256 threads = 8 waves; block covers a
// 16-row x 128-col tile, wave w owns column tile w*16. nrows % 16 == 0.
// ---------------------------------------------------------------------------
__global__ __launch_bounds__(256)
void gat_gemm128_wmma(const float* __restrict__ A, const float* __restrict__ B,
                      float* __restrict__ C) {
  const int rowTile = blockIdx.x;
  const int wave = threadIdx.x >> 5;
  const int lane = threadIdx.x & 31;
  const int mr  = lane & 15;           // A row (in tile) / B col (in tile) this lane feeds
  const int ko  = (lane >> 4) << 1;    // K sub-offset {0,2} per half-wave
  const float* __restrict__ arow = A + (size_t)(rowTile * 16 + mr) * HC;
  const int ncol = wave * 16 + mr;

  v8f acc = {};
#pragma unroll
  for (int kb = 0; kb < HC; kb += 4) {
    v2f a, b;
    a.x = arow[kb + ko];
    a.y = arow[kb + ko + 1];
    b.x = B[(size_t)(kb + ko) * HC + ncol];
    b.y = B[(size_t)(kb + ko + 1) * HC + ncol];
    acc = __builtin_amdgcn_wmma_f32_16x16x4_f32(
        /*neg_a=*/false, a, /*neg_b=*/false, b,
        /*c_mod=*/(short)0, acc, /*reuse_a=*/false, /*reuse_b=*/false);
  }

  // C/D layout: VGPR v -> row (v + (lane/16)*8), col (lane&15) of the tile
  const int rbase = rowTile * 16 + ((lane >> 4) << 3);
  const int ccol  = wave * 16 + (lane & 15);
#pragma unroll
  for (int v = 0; v < 8; ++v) {
    C[(size_t)(rbase + v) * HC + ccol] = acc[v];
  }
}

// alpha_s/alpha_d[n,h] = dot(H[n, h*16:(h+1)*16], a_src/a_dst[h])
__global__ void gat_alpha(const float* __restrict__ H,
                          const float* __restrict__ a_src,
                          const float* __restrict__ a_dst,
                          float* __restrict__ as_, float* __restrict__ ad_, int n) {
  int idx = blockIdx.x * blockDim.x + threadIdx.x;
  if (idx >= n * HEADS) return;
  int node = idx >> 3, h = idx & 7;
  const float* hp = H + (size_t)node * HC + h * HIDDEN;
  float s = 0.f, d = 0.f;
#pragma unroll
  for (int i = 0; i < HIDDEN; ++i) {
    float v = hp[i];
    s += v * a_src[h * HIDDEN + i];
    d += v * a_dst[h * HIDDEN + i];
  }
  as_[idx] = s;
  ad_[idx] = d;
}

// F[n,c] = bias[c]; mbuf[n,h] = -inf; denom[n,h] = 0
__global__ void gat_init(float* __restrict__ F, const float* __restrict__ bias,
                         float* __restrict__ mbuf, float* __restrict__ denom, int n) {
  int idx = blockIdx.x * blockDim.x + threadIdx.x;
  if (idx < n * HC)    F[idx] = bias[idx & (HC - 1)];
  if (idx < n * HEADS) { mbuf[idx] = -__builtin_inff(); denom[idx] = 0.f; }
}

__device__ __forceinline__ void edge_sd(const long long* __restrict__ ei,
                                        int e, int E, int& s, int& d) {
  if (e < E) { s = (int)ei[e]; d = (int)ei[E + e]; }
  else       { s = d = e - E; }   // self loops appended
}

// pass 1: segment max of leaky_relu(alpha_s[src]+alpha_d[dst]) by dst
__global__ void gat_edge_max(const long long* __restrict__ ei,
                             const float* __restrict__ as_, const float* __restrict__ ad_,
                             float* __restrict__ mbuf, int E, int E2) {
  int e = blockIdx.x * blockDim.x + threadIdx.x;
  if (e >= E2) return;
  int s, d; edge_sd(ei, e, E, s, d);
#pragma unroll
  for (int h = 0; h < HEADS; ++h) {
    float v = as_[s * HEADS + h] + ad_[d * HEADS + h];
    v = v > 0.f ? v : NEG_SLOPE * v;
    atomicMax(&mbuf[d * HEADS + h], v);   // global_atomic_max_num_f32
  }
}

// pass 2: denom[dst,h] += exp(e - m[dst,h])
__global__ void gat_edge_sum(const long long* __restrict__ ei,
                             const float* __restrict__ as_, const float* __restrict__ ad_,
                             const float* __restrict__ mbuf, float* __restrict__ denom,
                             int E, int E2) {
  int e = blockIdx.x * blockDim.x + threadIdx.x;
  if (e >= E2) return;
  int s, d; edge_sd(ei, e, E, s, d);
#pragma unroll
  for (int h = 0; h < HEADS; ++h) {
    float v = as_[s * HEADS + h] + ad_[d * HEADS + h];
    v = v > 0.f ? v : NEG_SLOPE * v;
    atomicAdd(&denom[d * HEADS + h], __expf(v - mbuf[d * HEADS + h]));
  }
}

// pass 3: F[dst, h*16+c] += alpha * H[src, h*16+c]; one thread per (edge, head)
__global__ void gat_edge_aggr(const long long* __restrict__ ei,
                              const float* __restrict__ as_, const float* __restrict__ ad_,
                              const float* __restrict__ mbuf, const float* __restrict__ denom,
                              const float* __restrict__ H, float* __restrict__ F,
                              int E, int E2) {
  int idx = blockIdx.x * blockDim.x + threadIdx.x;
  if (idx >= E2 * HEADS) return;
  int e = idx >> 3, h = idx & 7;
  int s, d; edge_sd(ei, e, E, s, d);
  float v = as_[s * HEADS + h] + ad_[d * HEADS + h];
  v = v > 0.f ? v : NEG_SLOPE * v;
  float alpha = __expf(v - mbuf[d * HEADS + h]) / denom[d * HEADS + h];
  const float* hs = H + (size_t)s * HC + h * HIDDEN;
  float* fd = F + (size_t)d * HC + h * HIDDEN;
#pragma unroll
  for (int c = 0; c < HIDDEN; ++c) {
    atomicAdd(&fd[c], alpha * hs[c]);     // global_atomic_add_f32
  }
}

__global__ void gat_relu(float* __restrict__ F, int total) {
  int idx = blockIdx.x * blockDim.x + threadIdx.x;
  if (idx < total) F[idx] = fmaxf(F[idx], 0.f);
}

// out[n, j] = bout[j] + dot(F[n,:], Wout[:, j]),  Wout is [128 x 10]
__global__ void gat_out_proj(const float* __restrict__ F, const float* __restrict__ Wout,
                             const float* __restrict__ bout, float* __restrict__ out,
                             int n, int ncls) {
  int idx = blockIdx.x * blockDim.x + threadIdx.x;
  if (idx >= n * ncls) return;
  int node = idx / ncls, j = idx % ncls;
  const float* fr = F + (size_t)node * HC;
  float s = bout[j];
#pragma unroll 8
  for (int k = 0; k < HC; ++k) s += fr[k] * Wout[k * ncls + j];
  out[idx] = s;
}

extern "C" void kernel_launch(void* const* d_in, const int* in_sizes, int n_in,
                              void* d_out, int out_size, void* d_ws, size_t ws_size,
                              hipStream_t stream) {
  const float*     x      = (const float*)d_in[0];
  const long long* ei     = (const long long*)d_in[1];
  const float*     W1     = (const float*)d_in[2];
  const float*     a_src1 = (const float*)d_in[3];
  const float*     a_dst1 = (const float*)d_in[4];
  const float*     b1     = (const float*)d_in[5];
  const float*     W2     = (const float*)d_in[6];
  const float*     a_src2 = (const float*)d_in[7];
  const float*     a_dst2 = (const float*)d_in[8];
  const float*     b2     = (const float*)d_in[9];
  const float*     Wout   = (const float*)d_in[10];
  const float*     bout   = (const float*)d_in[11];
  float*           out    = (float*)d_out;

  const int N  = in_sizes[0] / HC;       // 50000 (multiple of 16)
  const int E  = in_sizes[1] / 2;        // 800000
  const int E2 = E + N;                  // + self loops
  const int NCLS = out_size / N;         // 10

  // Workspace layout (floats)
  float* H   = (float*)d_ws;             // [N,128] gemm output
  float* F   = H   + (size_t)N * HC;     // [N,128] aggregated features (init = bias)
  float* as_ = F   + (size_t)N * HC;     // [N,8]
  float* ad_ = as_ + (size_t)N * HEADS;  // [N,8]
  float* mb  = ad_ + (size_t)N * HEADS;  // [N,8] segment max
  float* dn  = mb  + (size_t)N * HEADS;  // [N,8] segment sum

  const int T = 256;
  dim3 gN128((N * HC + T - 1) / T), gNH((N * HEADS + T - 1) / T);
  dim3 gE((E2 + T - 1) / T), gEH((E2 * HEADS + T - 1) / T);
  dim3 gGemm(N / 16);

  // ---- Layer 1 ----
  gat_gemm128_wmma<<<gGemm, T, 0, stream>>>(x, W1, H);
  gat_alpha<<<gNH, T, 0, stream>>>(H, a_src1, a_dst1, as_, ad_, N);
  gat_init<<<gN128, T, 0, stream>>>(F, b1, mb, dn, N);
  gat_edge_max<<<gE, T, 0, stream>>>(ei, as_, ad_, mb, E, E2);
  gat_edge_sum<<<gE, T, 0, stream>>>(ei, as_, ad_, mb, dn, E, E2);
  gat_edge_aggr<<<gEH, T, 0, stream>>>(ei, as_, ad_, mb, dn, H, F, E, E2);
  gat_relu<<<gN128, T, 0, stream>>>(F, N * HC);

  // ---- Layer 2 ----
  gat_gemm128_wmma<<<gGemm, T, 0, stream>>>(F, W2, H);
  gat_alpha<<<gNH, T, 0, stream>>>(H, a_src2, a_dst2, as_, ad_, N);
  gat_init<<<gN128, T, 0, stream>>>(F, b2, mb, dn, N);
  gat_edge_max<<<gE, T, 0, stream>>>(ei, as_, ad_, mb, E, E2);
  gat_edge_sum<<<gE, T, 0, stream>>>(ei, as_, ad_, mb, dn, E, E2);
  gat_edge_aggr<<<gEH, T, 0, stream>>>(ei, as_, ad_, mb, dn, H, F, E, E2);
  gat_relu<<<gN128, T, 0, stream>>>(F, N * HC);

  // ---- Output projection ----
  dim3 gOut((N * NCLS + T - 1) / T);
  gat_out_proj<<<gOut, T, 0, stream>>>(F, Wout, bout, out, N, NCLS);
}